// HetGraphLayer_8160437862809
// MI455X (gfx1250) — compile-verified
//
#include <hip/hip_runtime.h>
#include <hip/hip_bf16.h>
#include <stdint.h>

typedef __attribute__((ext_vector_type(2))) float v2f;
typedef __attribute__((ext_vector_type(8))) float v8f;

#define NNODES 100000
#define DIM    128
#define NREL   3
#define NEDGE  400000

// ---------------------------------------------------------------------------
// out[n][d] = mean_r b[r][d]  (bias-only init; scatter accumulates on top)
// ---------------------------------------------------------------------------
__global__ void het_init_out(float* __restrict__ out, const float* __restrict__ b,
                             int total) {
    int i = blockIdx.x * blockDim.x + threadIdx.x;
    if (i < total) {
        int d = i & (DIM - 1);
        out[i] = (b[d] + b[DIM + d] + b[2 * DIM + d]) * (1.0f / 3.0f);
    }
}

__global__ void het_zero_u32(unsigned* __restrict__ p, int n) {
    int i = blockIdx.x * blockDim.x + threadIdx.x;
    if (i < n) p[i] = 0u;
}

// ---------------------------------------------------------------------------
// Degree histogram via u32 atomics (edges are int64 in the reference)
// ---------------------------------------------------------------------------
__global__ void het_degrees(const long long* __restrict__ src,
                            const long long* __restrict__ dst,
                            unsigned* __restrict__ cnt_out,
                            unsigned* __restrict__ cnt_in, int nE) {
    int e = blockIdx.x * blockDim.x + threadIdx.x;
    if (e < nE) {
        atomicAdd(&cnt_out[(int)src[e]], 1u);
        atomicAdd(&cnt_in[(int)dst[e]], 1u);
    }
}

__global__ void het_rsqrt_deg(const unsigned* __restrict__ cnt_out,
                              const unsigned* __restrict__ cnt_in,
                              float* __restrict__ rs_out,
                              float* __restrict__ rs_in, int n) {
    int i = blockIdx.x * blockDim.x + threadIdx.x;
    if (i < n) {
        rs_out[i] = rsqrtf(fmaxf((float)cnt_out[i], 1.0f));
        rs_in[i]  = rsqrtf(fmaxf((float)cnt_in[i], 1.0f));
    }
}

// ---------------------------------------------------------------------------
// h = (x * rs_out[:,None]) @ W   via V_WMMA_F32_16X16X4_F32.
// Block = 128 threads = 4 waves; each wave computes a 16x128 output strip
// (8 accumulator tiles of 16x16, K-loop of 32 steps of K=4 -> 256 WMMAs/wave).
// W (128x128 f32, 64KB) is staged transposed in LDS so B-fragments are
// contiguous ds_load_b64; A-fragments are contiguous global float2 loads.
//
// f32 16x4 A layout (ISA 7.12.2): lane l<16 holds A[m0+l][k],A[m0+l][k+1];
// lane l>=16 holds A[m0+(l-16)][k+2],[k+3]. B 4x16 mirrors with N=lane%16.
// C/D 16x16: VGPR j -> M = j + 8*(lane>=16), N = lane%16.
// ---------------------------------------------------------------------------
__global__ __launch_bounds__(128) void het_gemm_wmma(
    const float* __restrict__ x, const float* __restrict__ W,
    const float* __restrict__ rs_out, float* __restrict__ h, int nNodes) {
    __shared__ float Wt[DIM * DIM];  // Wt[col*128 + k] = W[k*128 + col]

    int tid = threadIdx.x;
    for (int i = tid; i < DIM * DIM; i += 128) {
        int kk = i >> 7;
        int col = i & (DIM - 1);
        Wt[col * DIM + kk] = W[i];
    }
    __syncthreads();

    int lane = tid & 31;
    int wave = tid >> 5;
    int m0   = blockIdx.x * 64 + wave * 16;

    int rowRaw = m0 + (lane & 15);
    int row    = rowRaw < nNodes ? rowRaw : (nNodes - 1);  // clamp, no EXEC change
    float s    = rs_out[row];
    int kofs   = (lane >> 4) << 1;                          // 0 or 2
    const float* xrow = x + (size_t)row * DIM + kofs;
    int colL = lane & 15;

    v8f c[8] = {};

    for (int k = 0; k < DIM; k += 4) {
        v2f a = *(const v2f*)(xrow + k);
        a *= s;
#pragma unroll
        for (int t = 0; t < 8; ++t) {
            const float* bp = Wt + ((t * 16 + colL) * DIM) + kofs + k;
            v2f bf = *(const v2f*)bp;
            c[t] = __builtin_amdgcn_wmma_f32_16x16x4_f32(
                false, a, false, bf, (short)0, c[t], false, false);
        }
    }

    int rowBase = m0 + ((lane >> 4) << 3);  // +8 for upper half-wave
#pragma unroll
    for (int t = 0; t < 8; ++t) {
#pragma unroll
        for (int j = 0; j < 8; ++j) {
            int rr = rowBase + j;
            if (rr < nNodes) h[(size_t)rr * DIM + t * 16 + colL] = c[t][j];
        }
    }
}

// ---------------------------------------------------------------------------
// One wave per edge: out[dst][:] += h[src][:] * rs_in[dst] / 3  (f32 atomics,
// L2-resident -> near-peak BW). 32 lanes x float4 = 128 columns.
// ---------------------------------------------------------------------------
__global__ void het_scatter(const float* __restrict__ h,
                            const long long* __restrict__ src,
                            const long long* __restrict__ dst,
                            const float* __restrict__ rs_in,
                            float* __restrict__ out, int nE) {
    int gw   = (blockIdx.x * blockDim.x + threadIdx.x) >> 5;  // global wave id
    int lane = threadIdx.x & 31;
    if (gw >= nE) return;
    int sN = (int)src[gw];
    int dN = (int)dst[gw];
    float scale = rs_in[dN] * (1.0f / 3.0f);
    const float4* hv = (const float4*)(h + (size_t)sN * DIM);
    float4 v = hv[lane];
    float* o = out + (size_t)dN * DIM + lane * 4;
    atomicAdd(o + 0, v.x * scale);
    atomicAdd(o + 1, v.y * scale);
    atomicAdd(o + 2, v.z * scale);
    atomicAdd(o + 3, v.w * scale);
}

// ---------------------------------------------------------------------------
extern "C" void kernel_launch(void* const* d_in, const int* in_sizes, int n_in,
                              void* d_out, int out_size, void* d_ws, size_t ws_size,
                              hipStream_t stream) {
    (void)in_sizes; (void)n_in; (void)out_size; (void)ws_size;

    const float*     x     = (const float*)d_in[0];
    const long long* edges = (const long long*)d_in[1];  // [3][2][400000] int64
    const float*     W     = (const float*)d_in[2];      // [3][128][128]
    const float*     b     = (const float*)d_in[3];      // [3][128]
    float*           out   = (float*)d_out;

    // workspace layout
    size_t hBytes = (size_t)NNODES * DIM * sizeof(float);           // 51.2 MB
    float*    h       = (float*)d_ws;
    unsigned* cnt_out = (unsigned*)((char*)d_ws + hBytes);          // NNODES
    unsigned* cnt_in  = cnt_out + NNODES;                           // NNODES
    float*    rs_out  = (float*)(cnt_in + NNODES);                  // NNODES
    float*    rs_in   = rs_out + NNODES;                            // NNODES

    const int total = NNODES * DIM;
    het_init_out<<<(total + 255) / 256, 256, 0, stream>>>(out, b, total);

    for (int r = 0; r < NREL; ++r) {
        const long long* src = edges + (size_t)r * 2 * NEDGE;
        const long long* dst = src + NEDGE;
        const float*     Wr  = W + (size_t)r * DIM * DIM;

        het_zero_u32<<<(2 * NNODES + 255) / 256, 256, 0, stream>>>(cnt_out, 2 * NNODES);
        het_degrees<<<(NEDGE + 255) / 256, 256, 0, stream>>>(src, dst, cnt_out, cnt_in, NEDGE);
        het_rsqrt_deg<<<(NNODES + 255) / 256, 256, 0, stream>>>(cnt_out, cnt_in,
                                                                rs_out, rs_in, NNODES);
        het_gemm_wmma<<<(NNODES + 63) / 64, 128, 0, stream>>>(x, Wr, rs_out, h, NNODES);

        int scatterThreads = NEDGE * 32;
        het_scatter<<<(scatterThreads + 255) / 256, 256, 0, stream>>>(h, src, dst,
                                                                      rs_in, out, NEDGE);
    }
}